// Rasterizer_56401510531599
// MI455X (gfx1250) — compile-verified
//
#include <hip/hip_runtime.h>
#include <hip/hip_bf16.h>
#include <stdint.h>

// ---------------------------------------------------------------------------
// Rasterizer color interpolation for MI455X (gfx1250).
//
// Memory-bound problem (~256 MB streaming @ 23.3 TB/s ~= 11 us floor; ~92
// MFLOP total, so WMMA is inapplicable). Strategy:
//   Phase 1: compact per-(batch,triangle) corner colors into 64 B records in
//            d_ws (36 MB, L2-resident) -> 1 random cacheline per pixel,
//            one dependent-gather level instead of two.
//   Phase 2: quad-vectorized shade; streaming ids/bary staged to LDS with
//            CDNA5 async copies (ASYNCcnt), deep L2 prefetch via
//            global_prefetch_b8; NON-TEMPORAL b128 stores for the 128 MB of
//            write-once output so the record table stays resident in L2.
// ---------------------------------------------------------------------------

constexpr int kB  = 8;
constexpr int kH  = 1024;
constexpr int kW  = 1024;
constexpr int kV  = 35709;
constexpr int kT  = 70789;
constexpr int kHW = kH * kW;               // 1 << 20
constexpr int kNQ = (kB * kHW) / 4;        // 2,097,152 pixel quads
constexpr int BLOCK = 256;
constexpr int GRID  = 2048;                // 524,288 threads -> 4 quads/thread

static_assert(kHW == (1 << 20), "HW must be 2^20 for shift addressing");

constexpr size_t kRecFloats = 16;          // one 64 B record per (b, t)
constexpr size_t kRecBytes  = (size_t)kB * (size_t)kT * kRecFloats * sizeof(float);

// Native vector types (HIP float4 is a struct; these work with the
// nontemporal builtins and lower to b128 VMEM ops).
typedef float f32x4 __attribute__((ext_vector_type(4)));
typedef int   i32x4 __attribute__((ext_vector_type(4)));

#if defined(__gfx1250__)
#define USE_ASYNC_LDS 1
#else
#define USE_ASYNC_LDS 0
#endif

__device__ __forceinline__ float clamp01(float x) {
  return fminf(fmaxf(x, 0.0f), 1.0f);
}

#if USE_ASYNC_LDS
// Low 32 bits of a flat pointer into LDS are the LDS byte address
// (ISA 10.2: LDS aperture maps addr[31:0] directly).
__device__ __forceinline__ unsigned lds_off(const void* p) {
  return (unsigned)(uintptr_t)p;
}
// GLOBAL_LOAD_ASYNC_TO_LDS_B128 (ISA 15.18.3 op 98): VDST = LDS byte address
// VGPR, VADDR = 64-bit global address, no SADDR. Tracked with ASYNCcnt.
__device__ __forceinline__ void async_copy_b128(const void* gsrc, unsigned loff) {
  asm volatile("global_load_async_to_lds_b128 %0, %1, off"
               :: "v"(loff), "v"(gsrc)
               : "memory");
}
__device__ __forceinline__ void stage_tile(const int* __restrict__ ids,
                                           const float* __restrict__ bary,
                                           int q, const i32x4* sid, const f32x4* sbar) {
  async_copy_b128(ids + 4 * (size_t)q, lds_off(sid));
  const float* gb = bary + 12 * (size_t)q;
  const unsigned bo = lds_off(sbar);
  async_copy_b128(gb + 0, bo + 0);
  async_copy_b128(gb + 4, bo + 16);
  async_copy_b128(gb + 8, bo + 32);
}
#endif

// ---------------------------------------------------------------------------
// Phase 1: build per-(batch,triangle) corner-color records in d_ws.
// Record layout (floats): [c0.r c0.g c0.b c1.r | c1.g c1.b c2.r c2.g | c2.b x x x | pad]
// Regular (cached) stores: this table must land and stay in L2.
// ---------------------------------------------------------------------------
__global__ __launch_bounds__(BLOCK) void build_tricolor_kernel(
    const int* __restrict__ tris, const float* __restrict__ diffuse,
    float* __restrict__ rec)
{
  const int t = blockIdx.x * BLOCK + threadIdx.x;
  const int b = blockIdx.y;
  if (t >= kT) return;
  const int* tp = tris + 3 * (size_t)t;
  const float* cb = diffuse + (size_t)b * (3 * kV);
  const float* c0 = cb + 3 * (size_t)tp[0];
  const float* c1 = cb + 3 * (size_t)tp[1];
  const float* c2 = cb + 3 * (size_t)tp[2];
  float* r = rec + ((size_t)b * kT + t) * kRecFloats;
  f32x4 w0 = {c0[0], c0[1], c0[2], c1[0]};
  f32x4 w1 = {c1[1], c1[2], c2[0], c2[1]};
  f32x4 w2 = {c2[2], 0.0f, 0.0f, 0.0f};
  *(f32x4*)(r + 0) = w0;
  *(f32x4*)(r + 4) = w1;
  *(f32x4*)(r + 8) = w2;
}

// ---------------------------------------------------------------------------
// Per-pixel shading helpers.
// ---------------------------------------------------------------------------
__device__ __forceinline__ void shade_px_rec(
    const float* __restrict__ rec, int b, int t,
    float b0, float b1, float b2,
    float& R, float& G, float& Bl, float& A)
{
  const float* r = rec + ((size_t)b * kT + t) * kRecFloats;
  const f32x4 u0 = *(const f32x4*)(r);      // c0.r c0.g c0.b c1.r
  const f32x4 u1 = *(const f32x4*)(r + 4);  // c1.g c1.b c2.r c2.g
  const float u2 = r[8];                    // c2.b
  R  = clamp01(fmaf(b0, u0.x, fmaf(b1, u0.w, b2 * u1.z)));
  G  = clamp01(fmaf(b0, u0.y, fmaf(b1, u1.x, b2 * u1.w)));
  Bl = clamp01(fmaf(b0, u0.z, fmaf(b1, u1.y, b2 * u2)));
  A  = clamp01(2.0f * ((b0 + b1) + b2));
}

__device__ __forceinline__ void shade_px_direct(
    const int* __restrict__ tris, const float* __restrict__ diffuse,
    int b, int t, float b0, float b1, float b2,
    float& R, float& G, float& Bl, float& A)
{
  const int* tp = tris + 3 * (size_t)t;
  const float* cb = diffuse + (size_t)b * (3 * kV);
  const float* c0 = cb + 3 * (size_t)tp[0];
  const float* c1 = cb + 3 * (size_t)tp[1];
  const float* c2 = cb + 3 * (size_t)tp[2];
  R  = clamp01(fmaf(b0, c0[0], fmaf(b1, c1[0], b2 * c2[0])));
  G  = clamp01(fmaf(b0, c0[1], fmaf(b1, c1[1], b2 * c2[1])));
  Bl = clamp01(fmaf(b0, c0[2], fmaf(b1, c1[2], b2 * c2[2])));
  A  = clamp01(2.0f * ((b0 + b1) + b2));
}

// Non-temporal b128 stores: 128 MB write-once output must not evict the
// L2-resident record table.
__device__ __forceinline__ void store_quad(float* __restrict__ out, int b, int hw,
                                           f32x4 r4, f32x4 g4, f32x4 b4, f32x4 a4)
{
  float* p = out + ((size_t)(3 * b) * kHW + hw);
  __builtin_nontemporal_store(r4, (f32x4*)(p));
  __builtin_nontemporal_store(g4, (f32x4*)(p + kHW));
  __builtin_nontemporal_store(b4, (f32x4*)(p + 2 * kHW));
  __builtin_nontemporal_store(a4,
      (f32x4*)(out + (size_t)(3 * kB) * kHW + (size_t)b * kHW + hw));
}

template <bool UseRec>
__device__ __forceinline__ void compute_quad(
    int q, i32x4 t4, f32x4 w0, f32x4 w1, f32x4 w2,
    const float* __restrict__ diffuse, const int* __restrict__ tris,
    const float* __restrict__ rec, float* __restrict__ out)
{
  const int p  = q << 2;          // flat pixel index (< 2^23)
  const int b  = p >> 20;         // batch
  const int hw = p & (kHW - 1);   // position in plane (multiple of 4)
  const float w[12] = {w0.x, w0.y, w0.z, w0.w,
                       w1.x, w1.y, w1.z, w1.w,
                       w2.x, w2.y, w2.z, w2.w};
  const int ti[4] = {t4.x, t4.y, t4.z, t4.w};
  float R[4], G[4], Bl[4], A[4];
#pragma unroll
  for (int i = 0; i < 4; ++i) {
    if (UseRec)
      shade_px_rec(rec, b, ti[i], w[3 * i], w[3 * i + 1], w[3 * i + 2],
                   R[i], G[i], Bl[i], A[i]);
    else
      shade_px_direct(tris, diffuse, b, ti[i], w[3 * i], w[3 * i + 1], w[3 * i + 2],
                      R[i], G[i], Bl[i], A[i]);
  }
  f32x4 r4 = {R[0], R[1], R[2], R[3]};
  f32x4 g4 = {G[0], G[1], G[2], G[3]};
  f32x4 b4 = {Bl[0], Bl[1], Bl[2], Bl[3]};
  f32x4 a4 = {A[0], A[1], A[2], A[3]};
  store_quad(out, b, hw, r4, g4, b4, a4);
}

// ---------------------------------------------------------------------------
// Phase 2: main shade kernel. Grid-stride over pixel quads; streaming inputs
// double-buffered into LDS via CDNA5 async copies, gathers issued directly.
// ---------------------------------------------------------------------------
template <bool UseRec>
__global__ __launch_bounds__(BLOCK) void shade_kernel(
    const int* __restrict__ ids, const float* __restrict__ bary,
    const float* __restrict__ diffuse, const int* __restrict__ tris,
    const float* __restrict__ rec, float* __restrict__ out)
{
  const int stride = GRID * BLOCK;
  const int q0 = blockIdx.x * BLOCK + threadIdx.x;

#if USE_ASYNC_LDS
  __shared__ i32x4 s_ids[2][BLOCK];         //  8 KB: 4 ids per thread per stage
  __shared__ f32x4 s_bar[2][BLOCK * 3];     // 24 KB: 12 bary floats per thread
  const int lt = threadIdx.x;

  if (q0 < kNQ) stage_tile(ids, bary, q0, &s_ids[0][lt], &s_bar[0][3 * lt]);

  int s = 0;
  for (int q = q0; q < kNQ; q += stride, s ^= 1) {
    const int qn = q + stride;
    if (qn < kNQ) {
      // Stage next tile (L2 -> LDS) while this one is consumed.
      stage_tile(ids, bary, qn, &s_ids[s ^ 1][lt], &s_bar[s ^ 1][3 * lt]);
      // Pull the tile after next into L2 (HBM -> L2).
      const long q2 = (long)q + 2L * stride;
      if (q2 < kNQ) {
        __builtin_prefetch(ids + 4 * q2, 0, 0);
        __builtin_prefetch(bary + 12 * q2, 0, 0);
        __builtin_prefetch(bary + 12 * q2 + 8, 0, 0);
      }
      // 4 async ops outstanding for the next stage; current stage complete.
      asm volatile("s_wait_asynccnt 4" ::: "memory");
    } else {
      asm volatile("s_wait_asynccnt 0" ::: "memory");
    }
    const i32x4 t4 = s_ids[s][lt];
    const f32x4 w0 = s_bar[s][3 * lt + 0];
    const f32x4 w1 = s_bar[s][3 * lt + 1];
    const f32x4 w2 = s_bar[s][3 * lt + 2];
    compute_quad<UseRec>(q, t4, w0, w1, w2, diffuse, tris, rec, out);
  }
#else
  for (int q = q0; q < kNQ; q += stride) {
    const long qn = (long)q + stride;
    if (qn < kNQ) {
      __builtin_prefetch(ids + 4 * qn, 0, 0);
      __builtin_prefetch(bary + 12 * qn, 0, 0);
      __builtin_prefetch(bary + 12 * qn + 8, 0, 0);
    }
    const i32x4 t4 = __builtin_nontemporal_load((const i32x4*)(ids + 4 * (size_t)q));
    const f32x4* gb = (const f32x4*)(bary + 12 * (size_t)q);
    const f32x4 w0 = __builtin_nontemporal_load(gb + 0);
    const f32x4 w1 = __builtin_nontemporal_load(gb + 1);
    const f32x4 w2 = __builtin_nontemporal_load(gb + 2);
    compute_quad<UseRec>(q, t4, w0, w1, w2, diffuse, tris, rec, out);
  }
#endif
}

// ---------------------------------------------------------------------------
extern "C" void kernel_launch(void* const* d_in, const int* in_sizes, int n_in,
                              void* d_out, int out_size, void* d_ws, size_t ws_size,
                              hipStream_t stream)
{
  (void)in_sizes; (void)n_in; (void)out_size;
  const int*   ids  = (const int*)d_in[0];    // [B,H,W] int32
  const float* bary = (const float*)d_in[1];  // [B,H,W,3] f32
  const float* dif  = (const float*)d_in[2];  // [B,V,3] f32
  const int*   tris = (const int*)d_in[3];    // [T,3] int32
  float* out = (float*)d_out;                 // images [B,3,H,W] ++ alphas [B,1,H,W]

  if (ws_size >= kRecBytes) {
    float* rec = (float*)d_ws;
    dim3 pgrid((kT + BLOCK - 1) / BLOCK, kB, 1);
    build_tricolor_kernel<<<pgrid, BLOCK, 0, stream>>>(tris, dif, rec);
    shade_kernel<true><<<GRID, BLOCK, 0, stream>>>(ids, bary, dif, tris, rec, out);
  } else {
    shade_kernel<false><<<GRID, BLOCK, 0, stream>>>(ids, bary, dif, tris, nullptr, out);
  }
}